// WavyBert_11012296147561
// MI455X (gfx1250) — compile-verified
//
#include <hip/hip_runtime.h>
#include <math.h>

typedef __attribute__((ext_vector_type(2))) float v2f;
typedef __attribute__((ext_vector_type(8))) float v8f;

#define L_CAP 400
#define B_SZ  32
#define T_SZ  1500
#define H_SZ  768
#define HM1   767
#define P_MIXC 0.15f
#define M_TOT (B_SZ * L_CAP)   /* 12800 rows */
#define LDA   772              /* padded LDS row stride (floats) */

/* workspace layout in floats */
#define WS_CA    0
#define WS_CR    (WS_CA + B_SZ * L_CAP)
#define WS_FIRED (WS_CR + B_SZ * L_CAP)
#define WS_NHAT  (WS_FIRED + B_SZ * L_CAP)
#define WS_SA    (WS_NHAT + 64)

// ---------------------------------------------------------------------------
// Kernel 1: per-batch alpha normalization + scalar CIF recurrence -> coeffs
// ---------------------------------------------------------------------------
__global__ __launch_bounds__(256) void k_scan(const float* __restrict__ hs,
                                              const int* __restrict__ lengths,
                                              float* __restrict__ ws) {
    __shared__ float sig_early[L_CAP];
    __shared__ float red[256];
    const int b = blockIdx.x, tid = threadIdx.x;
    const float* base = hs + (size_t)b * T_SZ * H_SZ + (H_SZ - 1);
    float partial = 0.0f;
    for (int t = tid; t < T_SZ; t += 256) {
        float x = base[(size_t)t * H_SZ];
        float s = 1.0f / (1.0f + expf(-x));
        partial += s;
        if (t < L_CAP) sig_early[t] = s;
    }
    red[tid] = partial;
    __syncthreads();
    for (int off = 128; off > 0; off >>= 1) {
        if (tid < off) red[tid] += red[tid + off];
        __syncthreads();
    }
    if (tid == 0) {
        float denom = red[0];
        int len = lengths[b]; if (len > L_CAP) len = L_CAP;
        float tll = (float)len;
        float scale = tll / denom;
        float d = tll - denom;
        ws[WS_NHAT + b] = d * d;
        float* ca = ws + WS_CA + b * L_CAP;
        float* cr = ws + WS_CR + b * L_CAP;
        int*   fr = (int*)(ws + WS_FIRED) + b * L_CAP;
        float a_r = 0.0f;
        for (int t = 0; t < L_CAP; ++t) {
            float a     = sig_early[t] * scale;
            float a_a   = a + a_r;
            bool  fired = (a_a >= 1.0f);
            float rem   = 1.0f - a_r;
            float a_r_f = a - rem;
            ca[t] = fired ? rem : a;     /* s_a = s_r_prev + ca*h        */
            cr[t] = a_r_f;               /* if fired: s_r = cr*h         */
            fr[t] = fired ? 1 : 0;
            a_r   = fired ? a_r_f : a_a;
        }
    }
}

// ---------------------------------------------------------------------------
// Kernel 1b: n_hat finalize
// ---------------------------------------------------------------------------
__global__ void k_nhat(const float* __restrict__ ws, float* __restrict__ out_nhat) {
    if (threadIdx.x == 0) {
        float s = 0.0f;
        for (int b = 0; b < B_SZ; ++b) s += ws[WS_NHAT + b];
        *out_nhat = s;
    }
}

// ---------------------------------------------------------------------------
// Kernel 2: per-(b,h) CIF accumulation -> s_a (with zero pad column 767)
// ---------------------------------------------------------------------------
__global__ __launch_bounds__(768) void k_cif(const float* __restrict__ hs,
                                             const float* __restrict__ ws,
                                             float* __restrict__ sa) {
    __shared__ float lca[L_CAP];
    __shared__ float lcr[L_CAP];
    __shared__ int   lfr[L_CAP];
    const int b = blockIdx.x, tid = threadIdx.x;
    for (int t = tid; t < L_CAP; t += 768) {
        lca[t] = ws[WS_CA + b * L_CAP + t];
        lcr[t] = ws[WS_CR + b * L_CAP + t];
        lfr[t] = ((const int*)(ws + WS_FIRED))[b * L_CAP + t];
    }
    __syncthreads();
    const int h = tid;
    float* sab = sa + (size_t)b * L_CAP * H_SZ + h;
    if (h < HM1) {
        const float* hb = hs + (size_t)b * T_SZ * H_SZ + h;
        float s_r = 0.0f;
        for (int t = 0; t < L_CAP; ++t) {
            float hv  = hb[(size_t)t * H_SZ];
            float sav = fmaf(lca[t], hv, s_r);
            s_r = lfr[t] ? (lcr[t] * hv) : sav;
            sab[(size_t)t * H_SZ] = sav;
        }
    } else { /* h == 767: K-pad column for GEMM1 */
        for (int t = 0; t < L_CAP; ++t) sab[(size_t)t * H_SZ] = 0.0f;
    }
}

// ---------------------------------------------------------------------------
// Kernel 3: fused  relu(sa@W1+b1)@W2+b2  + mixing, fp32 WMMA 16x16x4
// 1 block = 16 output rows; 8 waves * 6 tiles of 16 cols = 768 cols
// ---------------------------------------------------------------------------
__global__ __launch_bounds__(256) void k_gemm_fused(
    const float* __restrict__ sa, const float* __restrict__ W1,
    const float* __restrict__ b1, const float* __restrict__ W2,
    const float* __restrict__ b2, const float* __restrict__ bert,
    const float* __restrict__ mixu, float* __restrict__ out_mix,
    float* __restrict__ out_fc) {
    __shared__ float As[16 * LDA];
    __shared__ float Inter[16 * LDA];
    __shared__ float rowmask[16];

    const int tid = threadIdx.x;
    const int m0  = blockIdx.x * 16;

    for (int idx = tid; idx < 16 * H_SZ; idx += 256) {
        int r = idx / H_SZ, c = idx - r * H_SZ;
        As[r * LDA + c] = sa[(size_t)(m0 + r) * H_SZ + c];
    }
    if (tid < 16) {
        rowmask[tid] = (mixu[m0 + tid] < P_MIXC) ? 1.0f : 0.0f;
    }
    __syncthreads();

    const int lane  = tid & 31;
    const int rr    = lane & 15;           /* N (and A-row) index inside tile */
    const int khalf = (lane >> 4) << 1;    /* lanes 16-31 carry K=k0+2,k0+3   */
    const int moff  = (lane >> 4) << 3;    /* C/D: upper half lanes -> M+8    */
    const int nbase = (tid >> 5) * 96;     /* 96 columns per wave             */

    v8f acc[6];
    #pragma unroll
    for (int i = 0; i < 6; ++i) acc[i] = (v8f){0,0,0,0,0,0,0,0};

    /* ---- GEMM1: K=768 (col 767 of As is zero, W1 row index clamped) ---- */
    for (int k0 = 0; k0 < H_SZ; k0 += 4) {
        const int kk = k0 + khalf;
        v2f a = *(const v2f*)&As[rr * LDA + kk];          /* ds_load_b64 */
        const int kb0 = (kk     > 766) ? 766 : kk;
        const int kb1 = (kk + 1 > 766) ? 766 : (kk + 1);
        const float* w1a = W1 + (size_t)kb0 * H_SZ;
        const float* w1b = W1 + (size_t)kb1 * H_SZ;
        #pragma unroll
        for (int nt = 0; nt < 6; ++nt) {
            const int n = nbase + nt * 16 + rr;
            v2f bf; bf.x = w1a[n]; bf.y = w1b[n];
            acc[nt] = __builtin_amdgcn_wmma_f32_16x16x4_f32(
                false, a, false, bf, (short)0, acc[nt], false, false);
        }
    }

    /* bias + relu -> Inter (LDS) */
    #pragma unroll
    for (int nt = 0; nt < 6; ++nt) {
        const int n = nbase + nt * 16 + rr;
        const float bias = b1[n];
        #pragma unroll
        for (int r = 0; r < 8; ++r) {
            float v = acc[nt][r] + bias;
            v = v > 0.0f ? v : 0.0f;
            Inter[(r + moff) * LDA + n] = v;
        }
    }
    __syncthreads();

    /* ---- GEMM2: K=768 full ---- */
    v8f acc2[6];
    #pragma unroll
    for (int i = 0; i < 6; ++i) acc2[i] = (v8f){0,0,0,0,0,0,0,0};

    for (int k0 = 0; k0 < H_SZ; k0 += 4) {
        const int kk = k0 + khalf;
        v2f a = *(const v2f*)&Inter[rr * LDA + kk];
        const float* w2a = W2 + (size_t)kk * H_SZ;
        #pragma unroll
        for (int nt = 0; nt < 6; ++nt) {
            const int n = nbase + nt * 16 + rr;
            v2f bf; bf.x = w2a[n]; bf.y = w2a[n + H_SZ];
            acc2[nt] = __builtin_amdgcn_wmma_f32_16x16x4_f32(
                false, a, false, bf, (short)0, acc2[nt], false, false);
        }
    }

    /* bias + fc store + mixing */
    #pragma unroll
    for (int nt = 0; nt < 6; ++nt) {
        const int n = nbase + nt * 16 + rr;
        const float bias = b2[n];
        #pragma unroll
        for (int r = 0; r < 8; ++r) {
            const int M = r + moff;
            const size_t o = (size_t)(m0 + M) * H_SZ + n;
            const float fc = acc2[nt][r] + bias;
            out_fc[o] = fc;
            const float msk = rowmask[M];
            out_mix[o] = (1.0f - msk) * fc + msk * bert[o];
        }
    }
}

// ---------------------------------------------------------------------------
extern "C" void kernel_launch(void* const* d_in, const int* in_sizes, int n_in,
                              void* d_out, int out_size, void* d_ws, size_t ws_size,
                              hipStream_t stream) {
    const float* hs      = (const float*)d_in[0];
    const int*   lengths = (const int*)d_in[1];
    const float* W1      = (const float*)d_in[2];
    const float* b1      = (const float*)d_in[3];
    const float* W2      = (const float*)d_in[4];
    const float* b2      = (const float*)d_in[5];
    const float* bert    = (const float*)d_in[6];
    const float* mixu    = (const float*)d_in[7];

    float* out      = (float*)d_out;
    float* out_mix  = out;
    float* out_fc   = out + (size_t)M_TOT * H_SZ;
    float* out_nhat = out + 2 * (size_t)M_TOT * H_SZ;
    float* ws       = (float*)d_ws;
    float* sa       = ws + WS_SA;

    k_scan<<<B_SZ, 256, 0, stream>>>(hs, lengths, ws);
    k_nhat<<<1, 32, 0, stream>>>(ws, out_nhat);
    k_cif<<<B_SZ, 768, 0, stream>>>(hs, ws, sa);
    k_gemm_fused<<<M_TOT / 16, 256, 0, stream>>>(sa, W1, b1, W2, b2, bert,
                                                 mixu, out_mix, out_fc);
}